// AlpacaImitation_15109694948022
// MI455X (gfx1250) — compile-verified
//
#include <hip/hip_runtime.h>
#include <math.h>

typedef __attribute__((ext_vector_type(2))) float v2f;
typedef __attribute__((ext_vector_type(4))) float v4f;
typedef __attribute__((ext_vector_type(8))) float v8f;

#define BATCH 1024
#define XD    32
#define UD    8
#define PHI_D 128
#define HID_D 128

// ---------------------------------------------------------------------------
// Kernel A: 4-layer MLP (Linear-Tanh x3, Linear) via V_WMMA_F32_16X16X4_F32.
// 4 waves per block; each block computes a 16-row batch tile, each wave owns
// 2 of the 8 N-tiles (N=16 each). LDS ping-pong between layers with padded
// row strides (36 / 132) so the 16 lanes of a half-group hit distinct banks.
//
// WMMA f32 16x16x4 fragment layouts (ISA 7.12.2):
//   A (16x4):  lane = g*16 + m (m=M row), vgpr v holds K = v + 2*g
//   B (4x16):  lane = g*16 + n,           vgpr v holds K = v + 2*g
//   C/D 16x16: lane = g*16 + n,           vgpr r holds M = r + 8*g
// ---------------------------------------------------------------------------

template <int K, bool ACT>
__device__ __forceinline__ void mlp_layer(const float* __restrict__ in, int instride,
                                          const float* __restrict__ W,
                                          const float* __restrict__ bias,
                                          float* __restrict__ out, int outstride,
                                          int g, int hl, int ntbase)
{
    constexpr int KS = K / 4;
    float a0[KS], a1[KS];
#pragma unroll
    for (int ks = 0; ks < KS; ++ks) {
        const int k0 = ks * 4 + 2 * g;
        a0[ks] = in[hl * instride + k0];
        a1[ks] = in[hl * instride + k0 + 1];
    }
#pragma unroll
    for (int nti = 0; nti < 2; ++nti) {
        const int n  = (ntbase + nti) * 16 + hl;
        const float bn = bias[n];
        v8f c = {bn, bn, bn, bn, bn, bn, bn, bn};
#pragma unroll
        for (int ks = 0; ks < KS; ++ks) {
            const int k0 = ks * 4 + 2 * g;
            v2f av = {a0[ks], a1[ks]};
            v2f bv = {W[k0 * HID_D + n], W[(k0 + 1) * HID_D + n]};
            c = __builtin_amdgcn_wmma_f32_16x16x4_f32(
                    /*neg_a=*/false, av, /*neg_b=*/false, bv,
                    /*c_mod=*/(short)0, c, /*reuse_a=*/false, /*reuse_b=*/false);
        }
#pragma unroll
        for (int r = 0; r < 8; ++r) {
            float vret = ACT ? tanhf(c[r]) : c[r];
            out[(r + 8 * g) * outstride + n] = vret;
        }
    }
}

__global__ __launch_bounds__(128) void mlp_phi_kernel(
    const float* __restrict__ x,
    const float* __restrict__ W1, const float* __restrict__ b1,
    const float* __restrict__ W2, const float* __restrict__ b2,
    const float* __restrict__ W3, const float* __restrict__ b3,
    const float* __restrict__ W4, const float* __restrict__ b4,
    float* __restrict__ phiv)
{
    __shared__ float xs[16 * 36];     // x tile, padded stride 36 (bank-conflict free)
    __shared__ float hA[16 * 132];    // hidden ping, padded stride 132
    __shared__ float hB[16 * 132];    // hidden pong

    const int t    = threadIdx.x;        // 0..127
    const int lid  = t & 31;             // lane within wave
    const int wave = t >> 5;             // 0..3 -> owns N-tiles 2*wave, 2*wave+1
    const int m0   = blockIdx.x * 16;    // batch row base
    const int g    = lid >> 4;
    const int hl   = lid & 15;
    const int ntb  = wave * 2;

    // stage x tile (16x32) into LDS, coalesced
    for (int i = t; i < 16 * 32; i += 128) {
        const int r = i >> 5, c = i & 31;
        xs[r * 36 + c] = x[(m0 + r) * XD + c];
    }
    __syncthreads();

    mlp_layer<32,  true >(xs, 36,  W1, b1, hA, 132, g, hl, ntb);
    __syncthreads();
    mlp_layer<128, true >(hA, 132, W2, b2, hB, 132, g, hl, ntb);
    __syncthreads();
    mlp_layer<128, true >(hB, 132, W3, b3, hA, 132, g, hl, ntb);
    __syncthreads();
    // final layer: no activation, write straight to global phiv (row stride 128)
    mlp_layer<128, false>(hA, 132, W4, b4, phiv + (size_t)m0 * PHI_D, PHI_D, g, hl, ntb);
}

// ---------------------------------------------------------------------------
// Kernel B: per (b,u) bilinear forms over the streamed 128x128 L tile.
//   mu      = sum_{p,q} Q[p]   * L[p][q] * phi[q]
//   sigfac  = 1 + sum_{p,q} phi[p] * L[p][q] * phi[q]
// 128 threads/block; each iteration the block reads 128 consecutive float4s
// (fully coalesced, non-temporal: L is 512 MB single-use >> 192 MB L2).
// Thread t covers q = 4*(t&31)..+3 of row p = 4*i + (t>>5).
// ---------------------------------------------------------------------------
__global__ __launch_bounds__(128) void post_kernel(
    const float* __restrict__ Q,
    const float* __restrict__ L,
    const float* __restrict__ phiv,
    const float* __restrict__ logSigEps,
    float* __restrict__ out)
{
    __shared__ float phis[PHI_D];
    __shared__ float Qs[PHI_D];
    __shared__ float red[2 * 128];

    const int t  = threadIdx.x;       // 0..127
    const int bu = blockIdx.x;        // b*8 + u
    const int u  = bu & 7;

    phis[t] = phiv[(size_t)(bu >> 3) * PHI_D + t];
    Qs[t]   = Q[(size_t)bu * PHI_D + t];
    __syncthreads();

    const int w  = t >> 5;            // which row-slice within the block
    const int qb = 4 * (t & 31);
    const v4f pq = {phis[qb], phis[qb + 1], phis[qb + 2], phis[qb + 3]};

    const v4f* __restrict__ Lp = (const v4f*)(L + (size_t)bu * (PHI_D * PHI_D));

    float muacc = 0.0f, pacc = 0.0f;
#pragma unroll 4
    for (int i = 0; i < 32; ++i) {
        const v4f lv = __builtin_nontemporal_load(&Lp[i * 128 + t]);
        const float s = lv.x * pq.x + lv.y * pq.y + lv.z * pq.z + lv.w * pq.w;
        const int p = i * 4 + w;
        muacc += Qs[p]   * s;
        pacc  += phis[p] * s;
    }

    red[t]       = muacc;
    red[128 + t] = pacc;
    __syncthreads();
#pragma unroll
    for (int st = 64; st > 0; st >>= 1) {
        if (t < st) {
            red[t]       += red[t + st];
            red[128 + t] += red[128 + t + st];
        }
        __syncthreads();
    }

    if (t == 0) {
        const float mu     = red[0];
        const float sigfac = 1.0f + red[128];
        const float se     = __expf(logSigEps[u]);
        out[bu] = mu;                                     // mu block: (B,U,1) flat
        float* sig = out + BATCH * UD + (size_t)bu * UD;  // sig block: (B,U,U) flat
#pragma unroll
        for (int v = 0; v < UD; ++v)
            sig[v] = (v == u) ? se * sigfac : 0.0f;
    }
}

extern "C" void kernel_launch(void* const* d_in, const int* in_sizes, int n_in,
                              void* d_out, int out_size, void* d_ws, size_t ws_size,
                              hipStream_t stream) {
    const float* x         = (const float*)d_in[0];
    const float* Q         = (const float*)d_in[1];
    const float* L         = (const float*)d_in[2];
    const float* W1        = (const float*)d_in[3];
    const float* b1        = (const float*)d_in[4];
    const float* W2        = (const float*)d_in[5];
    const float* b2        = (const float*)d_in[6];
    const float* W3        = (const float*)d_in[7];
    const float* b3        = (const float*)d_in[8];
    const float* W4        = (const float*)d_in[9];
    const float* b4        = (const float*)d_in[10];
    const float* logSigEps = (const float*)d_in[11];

    float* phiv = (float*)d_ws;   // B * PHI f32 = 512 KB scratch

    mlp_phi_kernel<<<BATCH / 16, 128, 0, stream>>>(x, W1, b1, W2, b2, W3, b3, W4, b4, phiv);
    post_kernel<<<BATCH * UD, 128, 0, stream>>>(Q, L, phiv, logSigEps, (float*)d_out);
}